// RoiPoolingRandom_25950192402927
// MI455X (gfx1250) — compile-verified
//
#include <hip/hip_runtime.h>
#include <hip/hip_bf16.h>

// ---------------------------------------------------------------------------
// RoiPoolingRandom on MI455X (gfx1250), compile-only tuned.
//
// Strategy: 2-D summed-area tables (SAT) so every (box, channel) pooled value
// is 4 L2-resident corner lookups instead of a ~4096-term masked dot product.
// The two prefix scans that build the SAT are expressed as fp32 GEMMs against
// triangular all-ones matrices and executed on the CDNA5 matrix cores with
// V_WMMA_F32_16X16X4_F32 (exact fp32 — matches the fp32 reference closely).
// ---------------------------------------------------------------------------

typedef float v2f __attribute__((ext_vector_type(2)));
typedef float v8f __attribute__((ext_vector_type(8)));

#define CCH   1024
#define HH    64
#define WW    64
#define HWPX  4096      // 64*64
#define NCLS  21
#define CAMCH 64        // 21 cnt planes + pad, then 21 value planes + pad

// ---------------------------------------------------------------------------
// Kernel 0: per-pixel CAM argmax -> 64-channel map buffer
//   camin[k][p]      = (argmax(cam[:,p]) == k) ? 1 : 0        (k < 21)
//   camin[32+k][p]   = (argmax == k) ? cam[k,p] : 0           (k < 21)
//   padding channels zeroed.
// Coalesced: for fixed k all threads write consecutive p.
// ---------------------------------------------------------------------------
__global__ void cam_maps_kernel(const float* __restrict__ cam,
                                float* __restrict__ camin) {
    int p = blockIdx.x * blockDim.x + threadIdx.x;
    if (p >= HWPX) return;
    float best = cam[p];
    int   bi   = 0;
    for (int k = 1; k < NCLS; ++k) {
        float v = cam[k * HWPX + p];
        if (v > best) { best = v; bi = k; }   // strict > keeps first max (jnp.argmax)
    }
    for (int k = 0; k < NCLS; ++k) {
        float hit = (k == bi) ? 1.0f : 0.0f;
        camin[k * HWPX + p]        = hit;
        camin[(32 + k) * HWPX + p] = (k == bi) ? best : 0.0f;
    }
    for (int k = NCLS; k < 32; ++k) {
        camin[k * HWPX + p]        = 0.0f;
        camin[(32 + k) * HWPX + p] = 0.0f;
    }
}

// ---------------------------------------------------------------------------
// Kernel 1: x-prefix scan as GEMM  out = in * U   (U upper-triangular ones)
// Rows are independent length-64 scans (row = ch*64 + y).
// One wave handles a 16-row group; B (triangular) generated in registers.
// ---------------------------------------------------------------------------
__global__ void prefix_x_wmma(const float* __restrict__ in,
                              float* __restrict__ out, int rows) {
    int wave = threadIdx.x >> 5;
    int lane = threadIdx.x & 31;
    int rowBase = (blockIdx.x * 4 + wave) * 16;
    if (rowBase >= rows) return;              // wave-uniform: EXEC stays all-ones

    int l15  = lane & 15;
    int half = lane >> 4;                     // 0 or 1
    int arow = rowBase + l15;                 // A fragment: lane = M

    for (int j = 0; j < 4; ++j) {             // N tile (output x block)
        v8f acc = {0.f, 0.f, 0.f, 0.f, 0.f, 0.f, 0.f, 0.f};
        int n_g  = j * 16 + l15;              // B/D column (lane = N)
        int kmax = (j + 1) * 16;              // U is zero above column block j
        for (int kb = 0; kb < kmax; kb += 4) {
            int ka = kb + 2 * half;
            v2f a, b;
            const float* ap = in + (size_t)arow * WW + ka;
            a.x = ap[0];
            a.y = ap[1];
            b.x = (ka     <= n_g) ? 1.0f : 0.0f;   // U[k][n] = (k <= n)
            b.y = (ka + 1 <= n_g) ? 1.0f : 0.0f;
            acc = __builtin_amdgcn_wmma_f32_16x16x4_f32(
                      false, a, false, b, (short)0, acc, false, false);
        }
        for (int v = 0; v < 8; ++v) {
            int m = v + 8 * half;             // D: VGPR row -> M (+8 upper half)
            out[(size_t)(rowBase + m) * WW + n_g] = acc[v];
        }
    }
}

// ---------------------------------------------------------------------------
// Kernel 2: y-prefix scan as GEMM  S_ch = L * X_ch  (L lower-triangular ones)
// One wave per channel (64x64 matrix); A (triangular) generated in registers,
// B tiles loaded coalesced (lanes = consecutive x).
// ---------------------------------------------------------------------------
__global__ void prefix_y_wmma(const float* __restrict__ in,
                              float* __restrict__ out, int nch) {
    int wave = threadIdx.x >> 5;
    int lane = threadIdx.x & 31;
    int ch = blockIdx.x * 4 + wave;
    if (ch >= nch) return;                    // wave-uniform

    const float* X = in  + (size_t)ch * HWPX;
    float*       S = out + (size_t)ch * HWPX;
    int l15  = lane & 15;
    int half = lane >> 4;

    for (int m = 0; m < 4; ++m) {             // output y tile
        int y_g  = m * 16 + l15;              // A row (lane = M)
        int kmax = (m + 1) * 16;              // L is zero right of row block m
        for (int n = 0; n < 4; ++n) {         // output x tile
            v8f acc = {0.f, 0.f, 0.f, 0.f, 0.f, 0.f, 0.f, 0.f};
            int x_g = n * 16 + l15;           // B/D column (lane = N)
            for (int kb = 0; kb < kmax; kb += 4) {
                int ka = kb + 2 * half;
                v2f a, b;
                a.x = (ka     <= y_g) ? 1.0f : 0.0f;  // L[y][y'] = (y' <= y)
                a.y = (ka + 1 <= y_g) ? 1.0f : 0.0f;
                b.x = X[(size_t)ka * WW + x_g];
                b.y = X[(size_t)(ka + 1) * WW + x_g];
                acc = __builtin_amdgcn_wmma_f32_16x16x4_f32(
                          false, a, false, b, (short)0, acc, false, false);
            }
            for (int v = 0; v < 8; ++v) {
                int y = m * 16 + v + 8 * half;
                S[(size_t)y * WW + x_g] = acc[v];
            }
        }
    }
}

// ---------------------------------------------------------------------------
// Kernel 3: per-box O(1) SAT corner lookups.
// Block = one box. Threads stride channels for pool; threads 0..20 do the
// 21-class cnt/s stats; thread 0 reduces confidence + fg_score.
// All SAT data (16 MB + 1 MB) stays resident in the 192 MB L2.
// ---------------------------------------------------------------------------
__global__ void box_lookup_kernel(const float* __restrict__ SF,
                                  const float* __restrict__ SC,
                                  const int* __restrict__ boxes,
                                  const int* __restrict__ labels,
                                  float* __restrict__ pool,
                                  float* __restrict__ fg,
                                  float* __restrict__ conf) {
    int n = blockIdx.x;
    int x0  = boxes[n * 4 + 0];
    int y0  = boxes[n * 4 + 1];
    int x1e = boxes[n * 4 + 2];
    int y1e = boxes[n * 4 + 3];
    int x1 = x1e - 1, y1 = y1e - 1;
    int xa = x0 - 1,  ya = y0 - 1;
    float area     = (float)((x1e - x0) * (y1e - y0));
    float inv_area = 1.0f / area;

    bool hasL = (xa >= 0), hasT = (ya >= 0);
    int pDD = y1 * WW + x1;
    int pDL = hasL ? (y1 * WW + xa) : 0;
    int pTD = hasT ? (ya * WW + x1) : 0;
    int pTL = (hasL && hasT) ? (ya * WW + xa) : 0;
    float wDL = hasL ? -1.0f : 0.0f;
    float wTD = hasT ? -1.0f : 0.0f;
    float wTL = (hasL && hasT) ? 1.0f : 0.0f;

    for (int c = threadIdx.x; c < CCH; c += blockDim.x) {
        const float* Sc = SF + (size_t)c * HWPX;
        float s = Sc[pDD] + wDL * Sc[pDL] + wTD * Sc[pTD] + wTL * Sc[pTL];
        pool[(size_t)n * CCH + c] = s * inv_area;
    }

    __shared__ float sh_present[NCLS];
    __shared__ float sh_mean[NCLS];
    int t = threadIdx.x;
    if (t < NCLS) {
        const float* Cc = SC + (size_t)t * HWPX;
        float cnt = Cc[pDD] + wDL * Cc[pDL] + wTD * Cc[pTD] + wTL * Cc[pTL];
        const float* Vc = SC + (size_t)(32 + t) * HWPX;
        float sv  = Vc[pDD] + wDL * Vc[pDL] + wTD * Vc[pTD] + wTL * Vc[pTL];
        bool present = (cnt > 0.5f);          // cnt is an exact integer in fp32
        sh_present[t] = present ? 1.0f : 0.0f;
        sh_mean[t]    = present ? (sv / fmaxf(cnt, 1.0f)) : 0.0f;
    }
    __syncthreads();
    if (t == 0) {
        float np = 0.0f, sm = 0.0f;
        for (int k = 0; k < NCLS; ++k) { np += sh_present[k]; sm += sh_mean[k]; }
        conf[n] = sm / np;
        int lab = labels[n];
        const float* Cc = SC + (size_t)lab * HWPX;
        float cntl = Cc[pDD] + wDL * Cc[pDL] + wTD * Cc[pTD] + wTL * Cc[pTL];
        fg[n] = cntl * inv_area;
    }
}

// ---------------------------------------------------------------------------
// Fallback (only if d_ws is too small for the SATs): direct per-box loops.
// ---------------------------------------------------------------------------
__global__ void direct_box_kernel(const float* __restrict__ F,
                                  const float* __restrict__ CAM,
                                  const int* __restrict__ boxes,
                                  const int* __restrict__ labels,
                                  float* __restrict__ pool,
                                  float* __restrict__ fg,
                                  float* __restrict__ conf) {
    int n = blockIdx.x;
    int x0  = boxes[n * 4 + 0];
    int y0  = boxes[n * 4 + 1];
    int x1e = boxes[n * 4 + 2];
    int y1e = boxes[n * 4 + 3];
    float inv_area = 1.0f / (float)((x1e - x0) * (y1e - y0));

    for (int c = threadIdx.x; c < CCH; c += blockDim.x) {
        const float* Fc = F + (size_t)c * HWPX;
        float s = 0.0f;
        for (int y = y0; y < y1e; ++y)
            for (int x = x0; x < x1e; ++x)
                s += Fc[y * WW + x];
        pool[(size_t)n * CCH + c] = s * inv_area;
    }

    __shared__ float cnt[NCLS], sv[NCLS];
    int t = threadIdx.x;
    if (t < NCLS) {
        float c_ = 0.0f, s_ = 0.0f;
        for (int y = y0; y < y1e; ++y)
            for (int x = x0; x < x1e; ++x) {
                int p = y * WW + x;
                float best = CAM[p];
                int bi = 0;
                for (int k = 1; k < NCLS; ++k) {
                    float v = CAM[k * HWPX + p];
                    if (v > best) { best = v; bi = k; }
                }
                if (bi == t) { c_ += 1.0f; s_ += best; }
            }
        cnt[t] = c_; sv[t] = s_;
    }
    __syncthreads();
    if (t == 0) {
        float np = 0.0f, sm = 0.0f;
        for (int k = 0; k < NCLS; ++k)
            if (cnt[k] > 0.5f) { np += 1.0f; sm += sv[k] / cnt[k]; }
        conf[n] = sm / np;
        fg[n] = cnt[labels[n]] * inv_area;
    }
}

// ---------------------------------------------------------------------------
extern "C" void kernel_launch(void* const* d_in, const int* in_sizes, int n_in,
                              void* d_out, int out_size, void* d_ws, size_t ws_size,
                              hipStream_t stream) {
    const float* F      = (const float*)d_in[0];   // (1024, 64, 64) f32
    const float* CAM    = (const float*)d_in[1];   // (21, 64, 64)  f32
    const int*   boxes  = (const int*)d_in[2];     // (N, 4) i32
    const int*   labels = (const int*)d_in[3];     // (N,)   i32
    int nbox = in_sizes[3];

    float* pool = (float*)d_out;
    float* fg   = pool + (size_t)nbox * CCH;
    float* conf = fg + nbox;

    size_t needF = (size_t)CCH * HWPX;      // x-prefix of features
    size_t needS = (size_t)CCH * HWPX;      // feature SAT
    size_t needC = (size_t)CAMCH * HWPX;    // cam map / x-prefix / SAT (x3)
    size_t need_bytes = (needF + needS + 3 * needC) * sizeof(float);

    if (ws_size < need_bytes) {
        // Scratch too small: direct (slow but correct) path.
        direct_box_kernel<<<nbox, 256, 0, stream>>>(F, CAM, boxes, labels,
                                                    pool, fg, conf);
        return;
    }

    float* XF = (float*)d_ws;               // (1024, 64, 64) x-prefix
    float* SF = XF + needF;                 // (1024, 64, 64) full SAT
    float* CI = SF + needS;                 // (64, 64, 64) cam maps
    float* XC = CI + needC;                 // (64, 64, 64) cam x-prefix
    float* SC = XC + needC;                 // (64, 64, 64) cam SAT

    // Build CAM hit/value planes.
    cam_maps_kernel<<<HWPX / 256, 256, 0, stream>>>(CAM, CI);

    // Feature SAT: x-scan then y-scan, both on the WMMA pipe (fp32, exact-ish).
    prefix_x_wmma<<<(CCH * HH) / 64, 128, 0, stream>>>(F, XF, CCH * HH);
    prefix_y_wmma<<<CCH / 4, 128, 0, stream>>>(XF, SF, CCH);

    // CAM SATs (64 channels: 21 cnt + 21 value + padding).
    prefix_x_wmma<<<(CAMCH * HH) / 64, 128, 0, stream>>>(CI, XC, CAMCH * HH);
    prefix_y_wmma<<<CAMCH / 4, 128, 0, stream>>>(XC, SC, CAMCH);

    // O(1) corner lookups per (box, channel) + per-class stats.
    box_lookup_kernel<<<nbox, 256, 0, stream>>>(SF, SC, boxes, labels,
                                                pool, fg, conf);
}